// AffinityPropagate2_34265249088048
// MI455X (gfx1250) — compile-verified
//
#include <hip/hip_runtime.h>
#include <math.h>

// Problem constants (fixed by the reference harness).
#define BATCH   4
#define HEIGHT  352
#define WIDTH   1216
#define HW      (HEIGHT * WIDTH)     // 428032
#define NC      9
#define PROP_STEPS 8

// Tiling: 64x16 output tile, halo 2 (dilation-2 stencil), 256 threads (8 wave32).
#define TW    64
#define TH    16
#define HALO  2
#define LW    (TW + 2 * HALO)        // 68
#define LH    (TH + 2 * HALO)        // 20
#define BLK   256

// ---------------------------------------------------------------------------
// Async global -> LDS tile stage (CDNA5 ASYNCcnt path).
// GVS mode: mem = SADDR(64-bit sgpr pair) + VADDR(32-bit byte offset).
// VDST vgpr carries the LDS byte offset (low 32 bits of the generic pointer).
// ---------------------------------------------------------------------------
__device__ __forceinline__ void load_tile_async(float* __restrict__ sm,
                                                const float* __restrict__ xb,
                                                int bx, int by, int tid) {
  unsigned lds_base = (unsigned)(size_t)(void*)sm;
  for (int i = tid; i < LH * LW; i += BLK) {
    int r  = i / LW;
    int c  = i - r * LW;
    int gy = by - HALO + r;
    int gx = bx - HALO + c;
    if ((unsigned)gy < (unsigned)HEIGHT && (unsigned)gx < (unsigned)WIDTH) {
      unsigned lds_addr = lds_base + (unsigned)i * 4u;
      unsigned goff     = (unsigned)(gy * WIDTH + gx) * 4u;
      asm volatile("global_load_async_to_lds_b32 %0, %1, %2"
                   :: "v"(lds_addr), "v"(goff), "s"(xb)
                   : "memory");
    } else {
      sm[i] = 0.0f;  // zero padding (reference pads with zeros); disjoint from async targets
    }
  }
  asm volatile("s_wait_asynccnt 0" ::: "memory");  // own wave's async LDS writes landed
  __syncthreads();                                  // (also waits DScnt for the zero-fills)
}

// ---------------------------------------------------------------------------
// One-time: softmax + fuse folding, repacked for wide loads.
//   wv  : per-pixel 64B record = { w1[0..7], w2[0..7] }   (4 x float4, aligned)
//   wsc : per-pixel float2     = { w1[8], w2[8] }
// where wX[k] = softmax(gX)[k] * fX.
// Grid: (HW/BLK, BATCH) -> no integer division for the batch index.
// ---------------------------------------------------------------------------
__global__ __launch_bounds__(BLK) void softmax_fuse_kernel(
    const float* __restrict__ g1, const float* __restrict__ g2,
    const float* __restrict__ fuse,
    float4* __restrict__ wv, float2* __restrict__ wsc) {
  const int p = blockIdx.x * BLK + threadIdx.x;   // pixel within plane (HW % BLK == 0)
  const int b = blockIdx.y;
  const size_t cbase = (size_t)b * NC * HW + (size_t)p;
  const float f1 = fuse[(size_t)b * 2 * HW + p];
  const float f2 = fuse[(size_t)b * 2 * HW + HW + p];

  float v1[NC], v2[NC];
  float m1 = -INFINITY, m2 = -INFINITY;
#pragma unroll
  for (int k = 0; k < NC; ++k) {
    v1[k] = g1[cbase + (size_t)k * HW]; m1 = fmaxf(m1, v1[k]);
    v2[k] = g2[cbase + (size_t)k * HW]; m2 = fmaxf(m2, v2[k]);
  }
  float s1 = 0.0f, s2 = 0.0f;
#pragma unroll
  for (int k = 0; k < NC; ++k) {
    v1[k] = __expf(v1[k] - m1); s1 += v1[k];
    v2[k] = __expf(v2[k] - m2); s2 += v2[k];
  }
  const float c1 = f1 / s1;
  const float c2 = f2 / s2;

  float4* rec = wv + (size_t)(b * HW + p) * 4;
  rec[0] = make_float4(v1[0] * c1, v1[1] * c1, v1[2] * c1, v1[3] * c1);
  rec[1] = make_float4(v1[4] * c1, v1[5] * c1, v1[6] * c1, v1[7] * c1);
  rec[2] = make_float4(v2[0] * c2, v2[1] * c2, v2[2] * c2, v2[3] * c2);
  rec[3] = make_float4(v2[4] * c2, v2[5] * c2, v2[6] * c2, v2[7] * c2);
  wsc[(size_t)b * HW + p] = make_float2(v1[8] * c1, v2[8] * c2);
}

// ---------------------------------------------------------------------------
// One propagation step.
//   x_out = sum_k w1_k * x[+d1_k] + sum_k w2_k * x[+d2_k]
// Weights: 4 x b128 + 1 x b64 per pixel (72B, L2-resident).
// x tile staged async into LDS (halo 2, zero padded).
// ---------------------------------------------------------------------------
__global__ __launch_bounds__(BLK) void prop_step_kernel(
    const float* __restrict__ xin,
    const float4* __restrict__ wv, const float2* __restrict__ wsc,
    float* __restrict__ xout) {
  __shared__ float sm[LH * LW];
  const int bx  = blockIdx.x * TW;
  const int by  = blockIdx.y * TH;
  const int b   = blockIdx.z;
  const int tid = threadIdx.x;
  const float* xb = xin + (size_t)b * HW;

  load_tile_async(sm, xb, bx, by, tid);

  const int lx  = tid & (TW - 1);
  const int ly0 = tid / TW;

#pragma unroll
  for (int j = 0; j < (TW * TH) / BLK; ++j) {
    const int ly = ly0 + j * (BLK / TW);
    const int gy = by + ly;
    const int gx = bx + lx;
    const int pix = gy * WIDTH + gx;

    const float4* rec = wv + (size_t)(b * HW + pix) * 4;
    const float4 A = rec[0];
    const float4 Bv = rec[1];
    const float4 C = rec[2];
    const float4 D = rec[3];
    const float2 S = wsc[(size_t)b * HW + pix];

    // dilation-1 taps: rows ly+1..ly+3, cols lx+1..lx+3 (tap order ky*3+kx)
    const float* r0 = &sm[(ly + 1) * LW + lx];
    const float* r1 = &sm[(ly + 2) * LW + lx];
    const float* r2 = &sm[(ly + 3) * LW + lx];
    float acc;
    acc = A.x * r0[1];
    acc = fmaf(A.y, r0[2], acc);
    acc = fmaf(A.z, r0[3], acc);
    acc = fmaf(A.w, r1[1], acc);
    acc = fmaf(Bv.x, r1[2], acc);
    acc = fmaf(Bv.y, r1[3], acc);
    acc = fmaf(Bv.z, r2[1], acc);
    acc = fmaf(Bv.w, r2[2], acc);
    acc = fmaf(S.x,  r2[3], acc);

    // dilation-2 taps: rows ly, ly+2, ly+4, cols lx, lx+2, lx+4
    const float* q0 = &sm[(ly + 0) * LW + lx];
    const float* q1 = r1 - 2;            // (ly + 2) * LW + lx - 2 ... careful: use fresh
    const float* q2 = &sm[(ly + 4) * LW + lx];
    const float* qm = &sm[(ly + 2) * LW + lx];
    acc = fmaf(C.x, q0[0], acc);
    acc = fmaf(C.y, q0[2], acc);
    acc = fmaf(C.z, q0[4], acc);
    acc = fmaf(C.w, qm[0], acc);
    acc = fmaf(D.x, qm[2], acc);
    acc = fmaf(D.y, qm[4], acc);
    acc = fmaf(D.z, q2[0], acc);
    acc = fmaf(D.w, q2[2], acc);
    acc = fmaf(S.y, q2[4], acc);
    (void)q1;

    xout[(size_t)b * HW + pix] = acc;
  }
}

// ---------------------------------------------------------------------------
// Fallback (if ws too small to hold precomputed weights): recompute softmax
// per pixel each step. Same async-LDS tile staging.
// ---------------------------------------------------------------------------
__global__ __launch_bounds__(BLK) void prop_step_fused_kernel(
    const float* __restrict__ xin,
    const float* __restrict__ g1, const float* __restrict__ g2,
    const float* __restrict__ fuse,
    float* __restrict__ xout) {
  __shared__ float sm[LH * LW];
  const int bx  = blockIdx.x * TW;
  const int by  = blockIdx.y * TH;
  const int b   = blockIdx.z;
  const int tid = threadIdx.x;
  const float* xb = xin + (size_t)b * HW;

  load_tile_async(sm, xb, bx, by, tid);

  const int lx  = tid & (TW - 1);
  const int ly0 = tid / TW;
  const size_t cb = (size_t)b * NC * HW;

#pragma unroll
  for (int j = 0; j < (TW * TH) / BLK; ++j) {
    const int ly = ly0 + j * (BLK / TW);
    const int gy = by + ly;
    const int gx = bx + lx;
    const size_t pix = (size_t)gy * WIDTH + gx;
    const float f1 = fuse[(size_t)b * 2 * HW + pix];
    const float f2 = fuse[(size_t)b * 2 * HW + HW + pix];

    float a[NC];
    float m = -INFINITY;
#pragma unroll
    for (int k = 0; k < NC; ++k) { a[k] = g1[cb + (size_t)k * HW + pix]; m = fmaxf(m, a[k]); }
    float s = 0.0f;
#pragma unroll
    for (int k = 0; k < NC; ++k) { a[k] = __expf(a[k] - m); s += a[k]; }
    float o1 = 0.0f;
#pragma unroll
    for (int ky = 0; ky < 3; ++ky)
#pragma unroll
      for (int kx = 0; kx < 3; ++kx)
        o1 = fmaf(a[ky * 3 + kx],
                  sm[(ly + HALO + (ky - 1)) * LW + (lx + HALO + (kx - 1))], o1);
    o1 /= s;

    m = -INFINITY;
#pragma unroll
    for (int k = 0; k < NC; ++k) { a[k] = g2[cb + (size_t)k * HW + pix]; m = fmaxf(m, a[k]); }
    s = 0.0f;
#pragma unroll
    for (int k = 0; k < NC; ++k) { a[k] = __expf(a[k] - m); s += a[k]; }
    float o2 = 0.0f;
#pragma unroll
    for (int ky = 0; ky < 3; ++ky)
#pragma unroll
      for (int kx = 0; kx < 3; ++kx)
        o2 = fmaf(a[ky * 3 + kx],
                  sm[(ly + HALO + 2 * (ky - 1)) * LW + (lx + HALO + 2 * (kx - 1))], o2);
    o2 /= s;

    xout[(size_t)b * HW + pix] = o1 * f1 + o2 * f2;
  }
}

// ---------------------------------------------------------------------------
extern "C" void kernel_launch(void* const* d_in, const int* in_sizes, int n_in,
                              void* d_out, int out_size, void* d_ws, size_t ws_size,
                              hipStream_t stream) {
  (void)in_sizes; (void)n_in; (void)out_size;
  const float* g1   = (const float*)d_in[0];
  const float* g2   = (const float*)d_in[1];
  const float* fuse = (const float*)d_in[2];
  const float* x0   = (const float*)d_in[3];
  float* out = (float*)d_out;

  const size_t planeBytes = (size_t)BATCH * HW * sizeof(float);           // ~6.85 MB (16B-mult.)
  const size_t wvBytes    = (size_t)BATCH * HW * 16 * sizeof(float);      // ~109.6 MB
  const size_t wscBytes   = (size_t)BATCH * HW * 2 * sizeof(float);       // ~13.7 MB

  dim3 grid(WIDTH / TW, HEIGHT / TH, BATCH);   // 19 x 22 x 4 (exact tiling)
  dim3 block(BLK);

  float* ping = (float*)d_ws;   // ping-pong buffer; steps with odd it (incl. the last, it=7) land in d_out

  if (ws_size >= planeBytes + wvBytes + wscBytes) {
    float4* wv  = (float4*)((char*)d_ws + planeBytes);
    float2* wsc = (float2*)((char*)d_ws + planeBytes + wvBytes);
    softmax_fuse_kernel<<<dim3(HW / BLK, BATCH), block, 0, stream>>>(g1, g2, fuse, wv, wsc);
    const float* src = x0;
    for (int it = 0; it < PROP_STEPS; ++it) {
      float* dst = (it & 1) ? out : ping;
      prop_step_kernel<<<grid, block, 0, stream>>>(src, wv, wsc, dst);
      src = dst;
    }
  } else {
    const float* src = x0;
    for (int it = 0; it < PROP_STEPS; ++it) {
      float* dst = (it & 1) ? out : ping;
      prop_step_fused_kernel<<<grid, block, 0, stream>>>(src, g1, g2, fuse, dst);
      src = dst;
    }
  }
}